// capsLayer_1915555414807
// MI455X (gfx1250) — compile-verified
//
#include <hip/hip_runtime.h>
#include <hip/hip_bf16.h>

// Capsule routing on MI455X (gfx1250), fp32 WMMA path.
// U is never materialized in HBM: B[b,c,r,o] == U[b,c,r,o]*A[b,c,o] with
// A = sum of past V's, so each iteration streams W (L2-resident, 75.5MB)
// and x, recomputes U tiles with V_WMMA_F32_16X16X4_F32, does the softmax
// over capsules in-register, and reduces S[b,c,o] with f32 global atomics.

#define Bt 32   // batch
#define Ct 32   // num capsules
#define Rt 4608 // route nodes
#define It 8    // in channels (K)
#define Ot 16   // out channels (N per capsule)
#define MB 16   // batch tile (WMMA M)
#define RB 32   // route nodes per block
#define LPAD 516 // padded LDS row stride per local-b (16*32 + 4)

typedef float v2f __attribute__((ext_vector_type(2)));
typedef float v8f __attribute__((ext_vector_type(8)));

__global__ __launch_bounds__(256) void caps_init_zero(float* p, int n) {
    int t = blockIdx.x * blockDim.x + threadIdx.x;
    if (t < n) p[t] = 0.0f;
}

__global__ __launch_bounds__(256) void caps_routing_pass(
    const float* __restrict__ x,   // [Bt][Rt][It]
    const float* __restrict__ W,   // [Ct][Rt][It][Ot]
    const float* __restrict__ A,   // [Bt][Ct][Ot]  (sum of past V)
    float* __restrict__ S)         // [Bt][Ct][Ot]  (zeroed before launch)
{
    __shared__ float Ulds[MB * LPAD]; // U tile [b_local][c*16+o], padded rows

    const int tid  = threadIdx.x;
    const int wave = tid >> 5;
    const int lane = tid & 31;
    const int b0   = blockIdx.y * MB;       // batch tile origin
    const int r0   = blockIdx.x * RB;       // route-node chunk origin

    // softmax-phase mapping: one thread per (local b, o)
    const int sb = tid >> 4;                // 0..15
    const int so = tid & 15;                // 0..15
    const int bg = b0 + sb;                 // global batch index

    // Per-thread row of A over capsules (constant during the r loop).
    float a_row[Ct];
    #pragma unroll
    for (int c = 0; c < Ct; ++c) a_row[c] = A[(bg * Ct + c) * Ot + so];

    float s_acc[Ct];
    #pragma unroll
    for (int c = 0; c < Ct; ++c) s_acc[c] = 0.0f;

    // WMMA fragment lane mapping (ISA 7.12.2, 32-bit 16x4 A / 4x16 B):
    // lanes 0-15 hold K={k0,k0+1}, lanes 16-31 hold K={k0+2,k0+3}.
    const int mrow  = lane & 15;            // M (local b) for A-frag; N (o) for B-frag
    const int khalf = (lane >> 4) * 2;      // K sub-offset inside a K=4 step
    const int dbase = (lane >> 4) * 8;      // D rows: j or j+8

    for (int rr = 0; rr < RB; ++rr) {
        const int r = r0 + rr;

        // ---- GEMM phase: U_r[b(16), c*16+o] via fp32 WMMA, 2 K-steps of 4 ----
        const float* xp = x + ((size_t)(b0 + mrow) * Rt + r) * It;
        v2f a0, a1;
        a0.x = xp[khalf + 0]; a0.y = xp[khalf + 1];
        a1.x = xp[4 + khalf + 0]; a1.y = xp[4 + khalf + 1];

        #pragma unroll
        for (int cc = 0; cc < 4; ++cc) {
            const int c = wave + cc * 8;    // this wave owns capsules w, w+8, w+16, w+24
            const float* wp = W + ((size_t)c * Rt + r) * (It * Ot);
            v2f bf0, bf1;
            bf0.x = wp[(khalf + 0) * Ot + mrow];
            bf0.y = wp[(khalf + 1) * Ot + mrow];
            bf1.x = wp[(4 + khalf + 0) * Ot + mrow];
            bf1.y = wp[(4 + khalf + 1) * Ot + mrow];

            v8f d = {};
            d = __builtin_amdgcn_wmma_f32_16x16x4_f32(false, a0, false, bf0,
                                                      (short)0, d, false, false);
            d = __builtin_amdgcn_wmma_f32_16x16x4_f32(false, a1, false, bf1,
                                                      (short)0, d, false, false);
            // D layout: VGPR j -> (M=j, N=lane) lanes0-15 ; (M=j+8, N=lane-16)
            #pragma unroll
            for (int j = 0; j < 8; ++j)
                Ulds[(dbase + j) * LPAD + c * Ot + mrow] = d[j];
        }
        __syncthreads();

        // ---- routing phase: softmax over c of (U*A), accumulate C*U into S ----
        float u[Ct], t[Ct];
        float m = -3.4e38f;
        #pragma unroll
        for (int c = 0; c < Ct; ++c) {
            u[c] = Ulds[sb * LPAD + c * Ot + so];
            t[c] = u[c] * a_row[c];
            m = fmaxf(m, t[c]);
        }
        float z = 0.0f;
        #pragma unroll
        for (int c = 0; c < Ct; ++c) { t[c] = __expf(t[c] - m); z += t[c]; }
        const float rz = __builtin_amdgcn_rcpf(z);
        #pragma unroll
        for (int c = 0; c < Ct; ++c) s_acc[c] += t[c] * rz * u[c];
        __syncthreads();
    }

    // Flush partial route-sums (one atomic per (b,c,o) per block).
    #pragma unroll
    for (int c = 0; c < Ct; ++c)
        atomicAdd(&S[(bg * Ct + c) * Ot + so], s_acc[c]);
}

// V = squash(S); A += V; S = 0 (ready for next pass); optionally emit V.
__global__ __launch_bounds__(256) void caps_squash_update(
    float* __restrict__ S, float* __restrict__ A,
    float* __restrict__ out, int write_out)
{
    int t = blockIdx.x * blockDim.x + threadIdx.x; // (b*Ct + c), 0..1023
    if (t >= Bt * Ct) return;
    float s[Ot];
    float sn = 0.0f;
    #pragma unroll
    for (int o = 0; o < Ot; ++o) { s[o] = S[t * Ot + o]; sn += s[o] * s[o]; }
    const float scale = sn / (1.0f + sn) * rsqrtf(sn);
    #pragma unroll
    for (int o = 0; o < Ot; ++o) {
        const float v = s[o] * scale;
        A[t * Ot + o] += v;
        if (write_out) out[t * Ot + o] = v;
        S[t * Ot + o] = 0.0f;
    }
}

extern "C" void kernel_launch(void* const* d_in, const int* in_sizes, int n_in,
                              void* d_out, int out_size, void* d_ws, size_t ws_size,
                              hipStream_t stream) {
    const float* x = (const float*)d_in[0];      // [32][4608][8]
    const float* W = (const float*)d_in[1];      // [32][4608][8][16]
    float* out = (float*)d_out;                  // [32][32][16]
    float* A = (float*)d_ws;                     // 16384 floats
    float* S = A + Bt * Ct * Ot;                 // 16384 floats

    const int zn = 2 * Bt * Ct * Ot;
    caps_init_zero<<<(zn + 255) / 256, 256, 0, stream>>>(A, zn);

    for (int it = 0; it < 3; ++it) {
        caps_routing_pass<<<dim3(Rt / RB, Bt / MB), 256, 0, stream>>>(x, W, A, S);
        caps_squash_update<<<dim3((Bt * Ct + 255) / 256), 256, 0, stream>>>(
            S, A, out, it == 2 ? 1 : 0);
    }
    (void)in_sizes; (void)n_in; (void)out_size; (void)ws_size;
}